// Unbounded_Metalog_Model_31860067401878
// MI455X (gfx1250) — compile-verified
//
#include <hip/hip_runtime.h>
#include <math.h>

#define TPB 256
#define PPT 4
#define PTS_PER_BLOCK (TPB * PPT)
#define ITERS 64
#define EPSV 1e-7f
#define LN2 0.69314718055994531f

// natural log via hardware v_log_f32 (log2)
__device__ __forceinline__ float hw_log2(float v) {
  return __builtin_amdgcn_logf(v);
}

// tanh(z) for z >= 0.  gfx1250 has V_TANH_F32; fall back to exp2-based form.
__device__ __forceinline__ float fast_tanh_pos(float z) {
#if __has_builtin(__builtin_amdgcn_tanhf)
  return __builtin_amdgcn_tanhf(z);
#else
  // tanh(z) = 1 - 2/(exp(2z)+1),  exp(2z) = exp2(2*log2(e)*z)
  float e = __builtin_amdgcn_exp2f(2.885390081777927f * z);
  return 1.0f - 2.0f * __builtin_amdgcn_rcpf(e + 1.0f);
#endif
}

// Grid: x-dim = blocks within one batch, y-dim = batch index (no int division).
__global__ __launch_bounds__(TPB) void metalog_pdf_kernel(
    const float* __restrict__ x, const float* __restrict__ a,
    float* __restrict__ out, int total, float adj0) {
  __shared__ float4 xs[TPB];

  const int tid   = threadIdx.x;
  const int blk   = blockIdx.y * gridDim.x + blockIdx.x;
  const int base  = blk * PTS_PER_BLOCK;
  const int idx0  = base + tid * PPT;

  // Coefficients: batch index is blockIdx.y -> pure scalar loads into SGPRs.
  const float* __restrict__ ac = a + blockIdx.y * 9;
  const float a0 = ac[0], a1 = ac[1], a2 = ac[2], a3 = ac[3], a4 = ac[4],
              a5 = ac[5], a6 = ac[6], a7 = ac[7], a8 = ac[8];

  float xv[PPT];
  const bool full = (base + PTS_PER_BLOCK) <= total;
  if (full) {
    // CDNA5 async data mover: 16B/lane global -> LDS (ASYNCcnt-tracked),
    // bypasses VGPR staging for the streamed input tile.
    unsigned lds_off = (unsigned)(size_t)(&xs[tid]);
    unsigned long long gaddr = (unsigned long long)(x + idx0);
    asm volatile("global_load_async_to_lds_b128 %0, %1, off"
                 :: "v"(lds_off), "v"(gaddr) : "memory");
    asm volatile("s_wait_asynccnt 0" ::: "memory");
    float4 v = xs[tid];
    xv[0] = v.x; xv[1] = v.y; xv[2] = v.z; xv[3] = v.w;
  } else {
#pragma unroll
    for (int i = 0; i < PPT; ++i) {
      int id = idx0 + i;
      xv[i] = (id < total) ? x[id] : 0.5f;
    }
  }

  float cy[PPT], adj[PPT];
#pragma unroll
  for (int i = 0; i < PPT; ++i) { cy[i] = 1.0f / 3.0f; adj[i] = adj0; }

  // 64-step fixed-point search. Iteration loop stays rolled (I$-friendly);
  // the PPT=4 independent chains inside give the ILP that hides trans latency.
#pragma unroll 1
  for (int it = 0; it < ITERS; ++it) {
#pragma unroll
    for (int i = 0; i < PPT; ++i) {
      float y = cy[i] + adj[i];
      cy[i] = y;
      float t = y - 0.5f;
      // L = ln(y/(1-y)) = ln2*(log2(y) - log2(1-y))   (2 trans ops)
      float L = LN2 * (hw_log2(y) - hw_log2(1.0f - y));
      // quantile = P1(t) + L*P2(t)
      float p1 = __builtin_fmaf(
          t, __builtin_fmaf(t, __builtin_fmaf(t, __builtin_fmaf(t, a8, a6), a4), a3), a0);
      float p2 = __builtin_fmaf(t, __builtin_fmaf(t, __builtin_fmaf(t, a7, a5), a2), a1);
      float q  = __builtin_fmaf(L, p2, p1);
      float diff = xv[i] - q;
      // heaviside(diff).clamp(eps, 1-eps) - cy
      float hv = (diff > 0.0f) ? (1.0f - EPSV) : EPSV;
      float th = fast_tanh_pos(diff * diff);            // 1 trans op
      adj[i] = (hv - y) * (th * (1.0f / 3.0f));
    }
  }

  // density = 1 / dquantile(cy);  dq = inv*P2(t) + L*P2'(t) + P1'(t)
#pragma unroll
  for (int i = 0; i < PPT; ++i) {
    int id = idx0 + i;
    if (id < total) {
      float y = cy[i];
      float t = y - 0.5f;
      float omy = 1.0f - y;
      float L  = LN2 * (hw_log2(y) - hw_log2(omy));
      float inv = 1.0f / (y * omy);
      float p2  = __builtin_fmaf(t, __builtin_fmaf(t, __builtin_fmaf(t, a7, a5), a2), a1);
      float dp2 = __builtin_fmaf(t, __builtin_fmaf(t, 3.0f * a7, 2.0f * a5), a2);
      float dp1 = __builtin_fmaf(
          t, __builtin_fmaf(t, __builtin_fmaf(t, 4.0f * a8, 3.0f * a6), 2.0f * a4), a3);
      float dq = __builtin_fmaf(inv, p2, __builtin_fmaf(L, dp2, dp1));
      float den = 1.0f / dq;
      // jnp.nan_to_num: nan->0, +/-inf -> +/-FLT_MAX
      if (__builtin_isnan(den)) den = 0.0f;
      else if (__builtin_isinf(den)) den = (den > 0.0f) ? 3.4028235e38f : -3.4028235e38f;
      out[id] = den;
    }
  }
}

extern "C" void kernel_launch(void* const* d_in, const int* in_sizes, int n_in,
                              void* d_out, int out_size, void* d_ws, size_t ws_size,
                              hipStream_t stream) {
  const float* x = (const float*)d_in[0];   // [B, P] float32
  const float* a = (const float*)d_in[1];   // [B, 9] float32
  float* out = (float*)d_out;               // [B, P] float32

  const int total = in_sizes[0];
  const int B = in_sizes[1] / 9;
  const int P = (B > 0) ? total / B : total;
  const int blocksPerBatch = (P + PTS_PER_BLOCK - 1) / PTS_PER_BLOCK;
  const float adj0 = 1.0f / (float)P;

  dim3 grid(blocksPerBatch, (B > 0) ? B : 1, 1);
  metalog_pdf_kernel<<<grid, TPB, 0, stream>>>(x, a, out, total, adj0);
}